// Mutcros_70411693850638
// MI455X (gfx1250) — compile-verified
//
#include <hip/hip_runtime.h>
#include <stdint.h>

// ---------------------------------------------------------------------------
// Mutcros on MI455X (gfx1250).
// Exact algebraic reduction: softmax rows sum to 1  =>  mean-pool of (v@att)
// == row-mean of v; mean over L commutes with conv1x1. Q/K branch is dead.
//   m      = mean_l cin                      [2][16(pad)][768]   (bandwidth)
//   pooled = m @ Wk^T + bk                   [2][16][768]        (WMMA f32)
//   mean   = relu(pooled @ Wo^T + bo) + og0
//   logvar = relu(pooled @ Wt^T + bt) + og1
//   out    = eps * exp(0.5*logvar) + mean    [2][8][768]
// ---------------------------------------------------------------------------

typedef __attribute__((ext_vector_type(2))) float v2f;
typedef __attribute__((ext_vector_type(8))) float v8f;

#define C_DIM 768
#define L_DIM 2048
#define B_DIM 8
#define MPAD  16   // pad batch 8 -> 16 for the WMMA M dimension

// ---------------- deterministic N(0,1) stand-in for jax threefry -----------
__device__ __forceinline__ uint32_t pcg_hash(uint32_t v) {
    uint32_t s = v * 747796405u + 2891336453u;
    uint32_t w = ((s >> ((s >> 28u) + 4u)) ^ s) * 277803737u;
    return (w >> 22u) ^ w;
}
__device__ __forceinline__ float gauss01(uint32_t seed) {
    uint32_t u1 = pcg_hash(seed ^ 0x68bc21ebu);
    uint32_t u2 = pcg_hash(seed ^ 0x02e5be93u);
    float f1 = ((float)u1 + 1.0f) * (1.0f / 4294967296.0f);  // (0,1]
    float f2 = (float)u2 * (1.0f / 4294967296.0f);           // [0,1)
    return sqrtf(-2.0f * logf(f1)) * __cosf(6.28318530718f * f2);
}

// ---------------- kernel 1: row means (bandwidth-bound, 100 MB) ------------
// One wave per [b,c] row of 2048 floats; float4 streaming loads.
// Rows b>=8 are zero padding for the WMMA A fragments.
__global__ __launch_bounds__(256) void mean_rows_kernel(
    const float* __restrict__ xcin, const float* __restrict__ ycin,
    float* __restrict__ m /* [2][MPAD][C] */) {
    const int wave = threadIdx.x >> 5;
    const int lane = threadIdx.x & 31;
    const int row  = blockIdx.x * 8 + wave;        // 0 .. 2*MPAD*C-1
    const int side = row / (MPAD * C_DIM);
    const int rem  = row % (MPAD * C_DIM);
    const int b    = rem / C_DIM;
    const int c    = rem % C_DIM;

    float sum = 0.0f;
    if (b < B_DIM) {
        const float* src = (side ? ycin : xcin) + (size_t)(b * C_DIM + c) * L_DIM;
        const float4* src4 = (const float4*)src;   // 512 float4 per row
#pragma unroll
        for (int i = 0; i < 16; ++i) {             // 16 b128 loads per lane
            float4 v = src4[lane + i * 32];
            sum += v.x + v.y + v.z + v.w;
        }
    }
#pragma unroll
    for (int off = 16; off > 0; off >>= 1) sum += __shfl_xor(sum, off, 32);
    if (lane == 0) m[row] = (b < B_DIM) ? sum * (1.0f / L_DIM) : 0.0f;
}

// A-fragment (16x4 f32): lanes 0-15 -> M=lane, K={k0,k0+1};
//                        lanes16-31 -> M=lane-16, K={k0+2,k0+3}.
// B-fragment (4x16 f32): lanes 0-15 -> N=n0+lane, K={k0,k0+1};
//                        lanes16-31 -> N=n0+lane-16, K={k0+2,k0+3}.
// D (16x16 f32): VGPR r, lanes0-15 -> row r, lanes16-31 -> row r+8; col=lane&15.

// ---------------- kernel 2: pooled = m @ Wk^T + bk (both sides) ------------
__global__ __launch_bounds__(32) void pooled_kernel(
    const float* __restrict__ m,     // [2][MPAD][C]
    const float* __restrict__ Wk,    // [C][C]
    const float* __restrict__ bk,    // [C]
    float* __restrict__ pooled) {    // [2][MPAD][C]
    const int lane = threadIdx.x;
    const int n0   = blockIdx.x * 16;
    const bool hi  = lane >= 16;
    const int l16  = lane & 15;
    const int ks   = hi ? 2 : 0;

    const float* mx   = m;
    const float* my   = m + MPAD * C_DIM;
    const float* wrow = Wk + (size_t)(n0 + l16) * C_DIM;  // B[k][n] = Wk[n][k]

    v8f accx = {}; v8f accy = {};
    for (int k0 = 0; k0 < C_DIM; k0 += 4) {
        v2f bf; bf.x = wrow[k0 + ks];             bf.y = wrow[k0 + ks + 1];
        v2f ax; ax.x = mx[l16 * C_DIM + k0 + ks]; ax.y = mx[l16 * C_DIM + k0 + ks + 1];
        v2f ay; ay.x = my[l16 * C_DIM + k0 + ks]; ay.y = my[l16 * C_DIM + k0 + ks + 1];
        accx = __builtin_amdgcn_wmma_f32_16x16x4_f32(false, ax, false, bf,
                                                     (short)0, accx, false, false);
        accy = __builtin_amdgcn_wmma_f32_16x16x4_f32(false, ay, false, bf,
                                                     (short)0, accy, false, false);
    }
    const int col   = n0 + l16;
    const float bia = bk[col];
    const int rbase = hi ? 8 : 0;
#pragma unroll
    for (int r = 0; r < 8; ++r) {
        pooled[(rbase + r) * C_DIM + col]                 = accx[r] + bia;
        pooled[MPAD * C_DIM + (rbase + r) * C_DIM + col]  = accy[r] + bia;
    }
}

// ---------------- kernel 3: head GEMMs + elementwise epilogue --------------
__global__ __launch_bounds__(32) void head_kernel(
    const float* __restrict__ pooled,  // [2][MPAD][C]
    const float* __restrict__ Wo, const float* __restrict__ bo,
    const float* __restrict__ Wt, const float* __restrict__ bt,
    const float* __restrict__ xog0, const float* __restrict__ xog1,
    const float* __restrict__ yog0, const float* __restrict__ yog1,
    float* __restrict__ out) {         // [2][B][C]
    const int lane = threadIdx.x;
    const int n0   = blockIdx.x * 16;
    const bool hi  = lane >= 16;
    const int l16  = lane & 15;
    const int ks   = hi ? 2 : 0;

    const float* px    = pooled;
    const float* py    = pooled + MPAD * C_DIM;
    const float* worow = Wo + (size_t)(n0 + l16) * C_DIM;
    const float* wtrow = Wt + (size_t)(n0 + l16) * C_DIM;

    v8f axo = {}; v8f axt = {}; v8f ayo = {}; v8f ayt = {};
    for (int k0 = 0; k0 < C_DIM; k0 += 4) {
        v2f bfo; bfo.x = worow[k0 + ks]; bfo.y = worow[k0 + ks + 1];
        v2f bft; bft.x = wtrow[k0 + ks]; bft.y = wtrow[k0 + ks + 1];
        v2f ax;  ax.x = px[l16 * C_DIM + k0 + ks]; ax.y = px[l16 * C_DIM + k0 + ks + 1];
        v2f ay;  ay.x = py[l16 * C_DIM + k0 + ks]; ay.y = py[l16 * C_DIM + k0 + ks + 1];
        axo = __builtin_amdgcn_wmma_f32_16x16x4_f32(false, ax, false, bfo,
                                                    (short)0, axo, false, false);
        axt = __builtin_amdgcn_wmma_f32_16x16x4_f32(false, ax, false, bft,
                                                    (short)0, axt, false, false);
        ayo = __builtin_amdgcn_wmma_f32_16x16x4_f32(false, ay, false, bfo,
                                                    (short)0, ayo, false, false);
        ayt = __builtin_amdgcn_wmma_f32_16x16x4_f32(false, ay, false, bft,
                                                    (short)0, ayt, false, false);
    }
    const int col    = n0 + l16;
    const float bso  = bo[col];
    const float bst  = bt[col];
    const int rbase  = hi ? 8 : 0;
#pragma unroll
    for (int r = 0; r < 8; ++r) {
        const int row = rbase + r;
        if (row < B_DIM) {
            const int idx = row * C_DIM + col;
            // x side
            float mnx = fmaxf(axo[r] + bso, 0.0f) + xog0[idx];
            float lvx = fmaxf(axt[r] + bst, 0.0f) + xog1[idx];
            float epx = 0.1f * gauss01((uint32_t)idx * 2654435761u + 0x1EC15EEDu);
            out[idx] = epx * __expf(0.5f * lvx) + mnx;
            // y side
            float mny = fmaxf(ayo[r] + bso, 0.0f) + yog0[idx];
            float lvy = fmaxf(ayt[r] + bst, 0.0f) + yog1[idx];
            float epy = 0.1f * gauss01((uint32_t)idx * 2654435761u + 0x9E3779B9u);
            out[B_DIM * C_DIM + idx] = epy * __expf(0.5f * lvy) + mny;
        }
    }
}

// ---------------------------------------------------------------------------
extern "C" void kernel_launch(void* const* d_in, const int* in_sizes, int n_in,
                              void* d_out, int out_size, void* d_ws, size_t ws_size,
                              hipStream_t stream) {
    const float* xcin = (const float*)d_in[0];
    const float* ycin = (const float*)d_in[1];
    const float* xog0 = (const float*)d_in[2];
    const float* xog1 = (const float*)d_in[3];
    const float* yog0 = (const float*)d_in[4];
    const float* yog1 = (const float*)d_in[5];
    // d_in[6]=Wq, d_in[7]=bq are provably dead (softmax rows sum to 1).
    const float* Wk = (const float*)d_in[8];
    const float* bk = (const float*)d_in[9];
    const float* Wo = (const float*)d_in[10];
    const float* bo = (const float*)d_in[11];
    const float* Wt = (const float*)d_in[12];
    const float* bt = (const float*)d_in[13];
    float* out = (float*)d_out;

    float* m      = (float*)d_ws;           // [2][16][768]
    float* pooled = m + 2 * MPAD * C_DIM;   // [2][16][768]

    // 2*16*768 rows / 8 waves per block = 3072 blocks
    mean_rows_kernel<<<3072, 256, 0, stream>>>(xcin, ycin, m);
    pooled_kernel<<<48, 32, 0, stream>>>(m, Wk, bk, pooled);
    head_kernel<<<48, 32, 0, stream>>>(pooled, Wo, bo, Wt, bt,
                                       xog0, xog1, yog0, yog1, out);
}